// BertCRFTagger_39943195852831
// MI455X (gfx1250) — compile-verified
//
#include <hip/hip_runtime.h>
#include <math.h>

typedef __attribute__((ext_vector_type(2))) float v2f;
typedef __attribute__((ext_vector_type(8))) float v8f;

#define Bz 32
#define Sz 512
#define Hz 1024
#define Tz 32

// ---------------------------------------------------------------------------
// Kernel 1: emissions = log_softmax(hidden @ W + b)  via V_WMMA_F32_16X16X4_F32
// One wave computes a 16-row x 32-col logits tile (two 16x16 accumulators).
// A layout (f32 16x4): lane = m (0..15) | +16 -> K pair {2,3}; VGPR0=K even, VGPR1=K odd.
// B layout (4x16):     lane = n (0..15) | +16 -> K pair {2,3}; VGPR0=K even, VGPR1=K odd.
// C/D layout: VGPR r, lanes 0-15 -> M=r, lanes 16-31 -> M=r+8, N=lane%16.
// ---------------------------------------------------------------------------
__global__ __launch_bounds__(256) void emis_kernel(const float* __restrict__ hidden,
                                                   const float* __restrict__ W,
                                                   const float* __restrict__ bias,
                                                   float* __restrict__ em) {
  const int lane = threadIdx.x & 31;
  const int wid  = threadIdx.x >> 5;
  const int tile = blockIdx.x * 8 + wid;        // 1024 tiles of 16 rows each
  const int m16  = lane & 15;
  const int hi   = lane >> 4;                   // 0: K={0,1}, 1: K={2,3}

  const float* arow = hidden + ((size_t)tile * 16 + m16) * Hz;

  v8f acc0 = {};
  v8f acc1 = {};
  #pragma unroll 4
  for (int k0 = 0; k0 < Hz; k0 += 4) {
    // A: 16x4 fp32 tile, 2 consecutive K values per lane (8B load)
    v2f a = *(const v2f*)(arow + k0 + 2 * hi);
    // B: 4x16 fp32 tiles from W (K-major, row stride Tz)
    const float* w0 = W + (size_t)(k0 + 2 * hi) * Tz;
    v2f b0, b1;
    b0.x = w0[m16];        b0.y = w0[Tz + m16];
    b1.x = w0[16 + m16];   b1.y = w0[Tz + 16 + m16];
    acc0 = __builtin_amdgcn_wmma_f32_16x16x4_f32(false, a, false, b0, (short)0, acc0, false, false);
    acc1 = __builtin_amdgcn_wmma_f32_16x16x4_f32(false, a, false, b1, (short)0, acc1, false, false);
  }

  const float bias0 = bias[m16];
  const float bias1 = bias[16 + m16];

  #pragma unroll
  for (int r = 0; r < 8; ++r) {
    float x0 = acc0[r] + bias0;               // N = m16
    float x1 = acc1[r] + bias1;               // N = m16+16
    // row-wise max over all 32 N values (xor masks <16 stay within lane halves)
    float mx = fmaxf(x0, x1);
    mx = fmaxf(mx, __shfl_xor(mx, 1));
    mx = fmaxf(mx, __shfl_xor(mx, 2));
    mx = fmaxf(mx, __shfl_xor(mx, 4));
    mx = fmaxf(mx, __shfl_xor(mx, 8));
    float se = __expf(x0 - mx) + __expf(x1 - mx);
    se += __shfl_xor(se, 1);
    se += __shfl_xor(se, 2);
    se += __shfl_xor(se, 4);
    se += __shfl_xor(se, 8);
    const float lse = mx + __logf(se);
    const size_t row = (size_t)tile * 16 + r + 8 * hi;   // output M
    em[row * Tz + m16]      = x0 - lse;
    em[row * Tz + 16 + m16] = x1 - lse;
  }
}

// ---------------------------------------------------------------------------
// Kernel 2: per-batch CRF numerator + forward (alpha) recursion. One wave/batch.
// Lane j owns state j. Per step: butterfly max, p=exp(alpha-mx), 32-term
// mat-vec against precomputed E[i][j]=exp(trans[i][j]) via readlane+fma.
// ---------------------------------------------------------------------------
__global__ __launch_bounds__(32) void crf_kernel(const float* __restrict__ em,
                                                 const float* __restrict__ start_trans,
                                                 const float* __restrict__ end_trans,
                                                 const float* __restrict__ trans,
                                                 const int*   __restrict__ tags,
                                                 const int*   __restrict__ mask,
                                                 float* __restrict__ llh_out) {
  const int b    = blockIdx.x;
  const int j    = threadIdx.x;          // 0..31  (state index)
  const int base = b * Sz;

  // exp() of transition column j (kept in 32 VGPRs per lane)
  float etr[Tz];
  #pragma unroll
  for (int i = 0; i < Tz; ++i) etr[i] = __expf(trans[i * Tz + j]);

  // ---- numerator partials + sequence length, lanes strided over S ----
  float numv = 0.0f;
  int   len  = 0;
  #pragma unroll
  for (int t = 0; t < Sz / 32; ++t) {
    const int s  = t * 32 + j;
    const int m  = mask[base + s];
    const int tg = tags[base + s];
    if (m) {
      numv += em[(size_t)(base + s) * Tz + tg];
      if (s > 0) {
        const int tp = tags[base + s - 1];
        numv += trans[tp * Tz + tg];
      }
      len += 1;
    }
  }
  #pragma unroll
  for (int d = 1; d < 32; d <<= 1) {
    numv += __shfl_xor(numv, d);
    len  += __shfl_xor(len, d);
  }
  numv += start_trans[tags[base]] + end_trans[tags[base + len - 1]];

  // ---- forward recursion over valid timesteps (mask is monotone) ----
  float alpha = start_trans[j] + em[(size_t)base * Tz + j];
  for (int s = 1; s < len; ++s) {
    float mx = alpha;
    mx = fmaxf(mx, __shfl_xor(mx, 1));
    mx = fmaxf(mx, __shfl_xor(mx, 2));
    mx = fmaxf(mx, __shfl_xor(mx, 4));
    mx = fmaxf(mx, __shfl_xor(mx, 8));
    mx = fmaxf(mx, __shfl_xor(mx, 16));
    const float p = __expf(alpha - mx);
    float a0 = 0.f, a1 = 0.f, a2 = 0.f, a3 = 0.f;
    #pragma unroll
    for (int i = 0; i < Tz; i += 4) {
      a0 = fmaf(__shfl(p, i + 0), etr[i + 0], a0);
      a1 = fmaf(__shfl(p, i + 1), etr[i + 1], a1);
      a2 = fmaf(__shfl(p, i + 2), etr[i + 2], a2);
      a3 = fmaf(__shfl(p, i + 3), etr[i + 3], a3);
    }
    alpha = em[(size_t)(base + s) * Tz + j] + mx + __logf((a0 + a1) + (a2 + a3));
  }

  // ---- denominator: logsumexp(alpha + end_trans) over all 32 lanes ----
  const float v = alpha + end_trans[j];
  float mx = v;
  #pragma unroll
  for (int d = 1; d < 32; d <<= 1) mx = fmaxf(mx, __shfl_xor(mx, d));
  float se = __expf(v - mx);
  #pragma unroll
  for (int d = 1; d < 32; d <<= 1) se += __shfl_xor(se, d);
  const float denom = mx + __logf(se);

  if (j == 0) llh_out[b] = numv - denom;
}

// ---------------------------------------------------------------------------
// Kernel 3: final reduce  out = -mean(llh)
// ---------------------------------------------------------------------------
__global__ __launch_bounds__(32) void reduce_kernel(const float* __restrict__ llh,
                                                    float* __restrict__ out) {
  float v = llh[threadIdx.x];
  #pragma unroll
  for (int d = 1; d < 32; d <<= 1) v += __shfl_xor(v, d);
  if (threadIdx.x == 0) out[0] = -v / (float)Bz;
}

extern "C" void kernel_launch(void* const* d_in, const int* in_sizes, int n_in,
                              void* d_out, int out_size, void* d_ws, size_t ws_size,
                              hipStream_t stream) {
  const float* hidden      = (const float*)d_in[0];   // (B,S,H)
  const float* W           = (const float*)d_in[1];   // (H,T)
  const float* bias        = (const float*)d_in[2];   // (T,)
  const float* start_trans = (const float*)d_in[3];   // (T,)
  const float* end_trans   = (const float*)d_in[4];   // (T,)
  const float* trans       = (const float*)d_in[5];   // (T,T)
  const int*   tags        = (const int*)d_in[6];     // (B,S)
  const int*   mask        = (const int*)d_in[7];     // (B,S)
  float* out = (float*)d_out;

  float* em  = (float*)d_ws;                          // B*S*T floats = 2 MB
  float* llh = em + (size_t)Bz * Sz * Tz;             // B floats

  // 16384 rows / 16 per wave = 1024 waves; 8 waves per 256-thread block
  emis_kernel<<<dim3(128), dim3(256), 0, stream>>>(hidden, W, bias, em);
  crf_kernel<<<dim3(Bz), dim3(32), 0, stream>>>(em, start_trans, end_trans, trans,
                                                tags, mask, llh);
  reduce_kernel<<<dim3(1), dim3(32), 0, stream>>>(llh, out);
}